// ExternalEmbeddingSelfAttention_22737556865750
// MI455X (gfx1250) — compile-verified
//
#include <hip/hip_runtime.h>
#include <hip/hip_bf16.h>

// ExternalEmbeddingSelfAttention for MI455X (gfx1250, wave32).
// - All GEMMs through V_WMMA_F32_16X16X4_F32 (fp32: unscaled softmax over
//   H=1024 dots makes bf16 downcast numerically unsafe).
// - Fused q/k/v projection (one A-fragment feeds 12 WMMAs per k-step).
// - W tiles staged to LDS by the Tensor Data Mover (tensor_load_to_lds),
//   double-buffered, synchronized with s_wait_tensorcnt; TDM padding
//   (4 dwords per 64) gives the bank-conflict-free LDS stride 68.

#define B_ 4
#define S_ 2048
#define E_ 512
#define H_ 1024

typedef __attribute__((ext_vector_type(2))) float v2f;
typedef __attribute__((ext_vector_type(8))) float v8f;
typedef __attribute__((ext_vector_type(4))) unsigned int u32x4;
typedef __attribute__((ext_vector_type(8))) int i32x8;
typedef __attribute__((ext_vector_type(4))) int i32x4;

#if __has_builtin(__builtin_amdgcn_tensor_load_to_lds) && \
    __has_builtin(__builtin_amdgcn_s_wait_tensorcnt)
#define USE_TDM 1
#else
#define USE_TDM 0
#endif

__device__ __forceinline__ v8f wmma_f32(v2f a, v2f b, v8f c) {
    // 8 args: (neg_a, A, neg_b, B, c_mod, C, reuse_a, reuse_b)
    return __builtin_amdgcn_wmma_f32_16x16x4_f32(false, a, false, b, (short)0, c,
                                                 false, false);
}

#if USE_TDM
// TDM: async copy of a 64x64 f32 tile from a row-major [*,1024] matrix into
// LDS, padding +4 dwords after every 64 dwords (row stride 68 floats).
// D# per CDNA5 ISA.
// clang-23 toolchain: 6-arg builtin (g0 u32x4, g1 i32x8, i32x4, i32x4, i32x8, cpol).
__device__ __forceinline__ void tdm_load_w64x64(const float* gsrc,
                                                const float* lds_dst) {
    unsigned long long ga = (unsigned long long)(uintptr_t)gsrc;
    unsigned int laddr   = (unsigned int)(uintptr_t)lds_dst; // LDS byte offset
    u32x4 g0;
    g0.x = 1u;                                    // count=1, is_restore=0
    g0.y = laddr;                                 // lds_addr
    g0.z = (unsigned int)ga;                      // global_addr[31:0]
    g0.w = (unsigned int)(ga >> 32) | (2u << 30); // global_addr[56:32] | type=2
    i32x8 g1;
    // data_size=4B(2), pad_enable, pad_interval=64dw(5), pad_amount=4dw(3)
    g1[0] = (int)((2u << 16) | (1u << 20) | (5u << 22) | (3u << 25));
    g1[1] = (int)((H_ & 0xFFFFu) << 16);          // tensor_dim0 lo16 @bits63:48
    g1[2] = (int)(((H_ >> 16) & 0xFFFFu) | ((H_ & 0xFFFFu) << 16)); // d0hi,d1lo
    g1[3] = (int)(((H_ >> 16) & 0xFFFFu) | (64u << 16)); // dim1 hi | tile_dim0
    g1[4] = 64;                                   // tile_dim1 (tile_dim2=0)
    g1[5] = (int)H_;                              // tensor_dim0_stride lo32
    g1[6] = 0;
    g1[7] = 0;
    i32x4 gz4 = {0, 0, 0, 0};                     // 2-D tensor: groups 2/3 off
    i32x8 gz8 = {0, 0, 0, 0, 0, 0, 0, 0};
    __builtin_amdgcn_tensor_load_to_lds(g0, g1, gz4, gz4, gz8, 0);
}
#endif

// ---------------------------------------------------------------------------
// Fused projections: C_o[M,H] = A[M,H] @ W_o[H,H] + b_o for o < NOUT.
// Block = 256 threads = 8 waves; block tile 128 rows x 64 cols.
// Each wave: 16 rows, NOUT*4 accumulators. W tiles double-buffered in LDS,
// filled by the TDM (wave 0 issues, all waves sync at barriers).
// ---------------------------------------------------------------------------
template <int NOUT>
__global__ __launch_bounds__(256) void proj_gemm_fused(
    const float* __restrict__ A,
    const float* __restrict__ W0, const float* __restrict__ W1,
    const float* __restrict__ W2,
    const float* __restrict__ b0, const float* __restrict__ b1,
    const float* __restrict__ b2,
    float* __restrict__ C0, float* __restrict__ C1, float* __restrict__ C2)
{
    constexpr int LDW  = 68;            // 64 + TDM pad (bank-conflict-free)
    constexpr int TILE = 64 * LDW;
    __shared__ float Wtile[2 * NOUT * TILE];

    const float* Ws[3]   = {W0, W1, W2};
    const float* bias[3] = {b0, b1, b2};
    float*       Cs[3]   = {C0, C1, C2};

    const int tid   = threadIdx.x;
    const int wave  = tid >> 5;
    const int lane  = tid & 31;
    const int row16 = lane & 15;
    const int hi    = lane >> 4;
    const int khalf = hi * 2;

    const int r0 = blockIdx.x * 128 + wave * 16;
    const int n0 = blockIdx.y * 64;

    const float* Arow = A + (size_t)(r0 + row16) * H_;

    v8f acc[NOUT][4] = {};

    constexpr int NSTAGE = H_ / 64;     // 16

#if USE_TDM
    if (wave == 0) {                    // preload stage 0 into buffer 0
        #pragma unroll
        for (int o = 0; o < NOUT; ++o)
            tdm_load_w64x64(Ws[o] + n0, &Wtile[o * TILE]);
    }
#endif

    for (int s = 0; s < NSTAGE; ++s) {
        const int cur = s & 1;
        const int k0  = s * 64;
#if USE_TDM
        if (wave == 0) __builtin_amdgcn_s_wait_tensorcnt(0);
        __syncthreads();                // buf[cur] ready; prev stage fully read
        if (wave == 0 && s + 1 < NSTAGE) {
            const int nxt = cur ^ 1;
            #pragma unroll
            for (int o = 0; o < NOUT; ++o)
                tdm_load_w64x64(Ws[o] + (size_t)(k0 + 64) * H_ + n0,
                                &Wtile[(nxt * NOUT + o) * TILE]);
        }
#else
        __syncthreads();
        for (int o = 0; o < NOUT; ++o)
            #pragma unroll
            for (int i = 0; i < 16; ++i) {
                int idx = tid + i * 256;
                int kk  = idx >> 6;
                int c   = idx & 63;
                Wtile[(cur * NOUT + o) * TILE + kk * LDW + c] =
                    Ws[o][(size_t)(k0 + kk) * H_ + n0 + c];
            }
        __syncthreads();
#endif
        #pragma unroll
        for (int kk = 0; kk < 64; kk += 4) {
            v2f a = *(const v2f*)(Arow + k0 + kk + khalf);
            #pragma unroll
            for (int o = 0; o < NOUT; ++o) {
                const float* tile = &Wtile[(cur * NOUT + o) * TILE];
                #pragma unroll
                for (int t = 0; t < 4; ++t) {
                    v2f b;
                    b.x = tile[(kk + khalf)     * LDW + t * 16 + row16];
                    b.y = tile[(kk + khalf + 1) * LDW + t * 16 + row16];
                    acc[o][t] = wmma_f32(a, b, acc[o][t]);
                }
            }
        }
        __syncthreads();                // all reads of buf[cur] done
    }

    #pragma unroll
    for (int o = 0; o < NOUT; ++o)
        #pragma unroll
        for (int t = 0; t < 4; ++t) {
            int col  = n0 + t * 16 + row16;
            float bv = bias[o][col];
            #pragma unroll
            for (int i = 0; i < 8; ++i) {
                int r = r0 + i + hi * 8;
                Cs[o][(size_t)r * H_ + col] = acc[o][t][i] + bv;
            }
        }
}

// ---------------------------------------------------------------------------
// Attention: one block per (batch, 16-row s-tile).  256 threads = 8 waves.
// ---------------------------------------------------------------------------
__global__ __launch_bounds__(256) void attn_kernel(
    const float* __restrict__ q,     const float* __restrict__ k_tok,
    const float* __restrict__ v_tok, const float* __restrict__ k_ext,
    const float* __restrict__ v_ext, float* __restrict__ out)
{
    constexpr int LDS_S = 516;              // 512 + pad
    __shared__ float Sc[16 * LDS_S];        // scores -> probs
    __shared__ float red[16 * 16];
    __shared__ float rowm[16];              // row max, then 1/denom
    __shared__ float sself[16];
    __shared__ float pself[16];

    const int b   = blockIdx.y;
    const int s0  = blockIdx.x * 16;
    const int tid = threadIdx.x;
    const int wave = tid >> 5, lane = tid & 31;
    const int row16 = lane & 15, hi = lane >> 4, khalf = hi * 2;

    const float* qb  = q     + ((size_t)b * S_ + s0) * H_;
    const float* keb = k_ext + (size_t)b * E_ * H_;
    const float* veb = v_ext + (size_t)b * E_ * H_;

    // ---- Stage 1: ext scores GEMM; wave w -> e in [w*64, w*64+64) ----
    {
        const int e0 = wave * 64;
        const float* Aq = qb + (size_t)row16 * H_;
        v8f acc[4] = {};
        for (int k0 = 0; k0 < H_; k0 += 4) {
            v2f a = *(const v2f*)(Aq + k0 + khalf);
            #pragma unroll
            for (int t = 0; t < 4; ++t) {
                v2f bb = *(const v2f*)(keb + (size_t)(e0 + t * 16 + row16) * H_
                                           + k0 + khalf);
                acc[t] = wmma_f32(a, bb, acc[t]);
            }
        }
        #pragma unroll
        for (int t = 0; t < 4; ++t)
            #pragma unroll
            for (int i = 0; i < 8; ++i)
                Sc[(i + hi * 8) * LDS_S + e0 + t * 16 + row16] = acc[t][i];
    }

    // ---- s_self[r] = dot(q[r], k_tok[r]) over H ----
    {
        const int r = tid & 15, j = tid >> 4;
        const float* qr = qb + (size_t)r * H_;
        const float* kr = k_tok + ((size_t)b * S_ + s0 + r) * H_;
        float p = 0.f;
        for (int h = j * 64; h < j * 64 + 64; h += 4) {
            float4 qv = *(const float4*)(qr + h);
            float4 kv = *(const float4*)(kr + h);
            p += qv.x * kv.x + qv.y * kv.y + qv.z * kv.z + qv.w * kv.w;
        }
        red[r * 16 + j] = p;
    }
    __syncthreads();
    if (tid < 16) {
        float s = 0.f;
        for (int j = 0; j < 16; ++j) s += red[tid * 16 + j];
        sself[tid] = s;
    }
    __syncthreads();

    // ---- softmax: row max ----
    {
        const int r = tid & 15, j = tid >> 4;
        float m = -3.4e38f;
        for (int e = j * 32; e < j * 32 + 32; ++e)
            m = fmaxf(m, Sc[r * LDS_S + e]);
        red[r * 16 + j] = m;
    }
    __syncthreads();
    if (tid < 16) {
        float m = sself[tid];
        for (int j = 0; j < 16; ++j) m = fmaxf(m, red[tid * 16 + j]);
        rowm[tid] = m;
    }
    __syncthreads();

    // ---- exp + partial sums ----
    {
        const int r = tid & 15, j = tid >> 4;
        float m = rowm[r], s = 0.f;
        for (int e = j * 32; e < j * 32 + 32; ++e) {
            float p = __expf(Sc[r * LDS_S + e] - m);
            Sc[r * LDS_S + e] = p;
            s += p;
        }
        red[r * 16 + j] = s;
    }
    __syncthreads();
    if (tid < 16) {
        float ps = __expf(sself[tid] - rowm[tid]);
        float d  = ps;
        for (int j = 0; j < 16; ++j) d += red[tid * 16 + j];
        float inv  = 1.f / d;
        rowm[tid]  = inv;
        pself[tid] = ps * inv;
    }
    __syncthreads();
    {   // normalize ext probs in LDS
        const int r = tid & 15, j = tid >> 4;
        float inv = rowm[r];
        for (int e = j * 32; e < j * 32 + 32; ++e)
            Sc[r * LDS_S + e] *= inv;
    }
    __syncthreads();

    // ---- Stage 3: ctx = P @ v_ext + p_self * v_tok; wave -> 128 h cols ----
    {
        const int n0 = wave * 128;
        v8f acc[8] = {};
        for (int k0 = 0; k0 < E_; k0 += 4) {
            const int kk = k0 + khalf;
            v2f a = *(const v2f*)(&Sc[row16 * LDS_S + kk]);   // ds_load_b64
            #pragma unroll
            for (int t = 0; t < 8; ++t) {
                v2f bb;
                bb.x = veb[(size_t)kk * H_ + n0 + t * 16 + row16];
                bb.y = veb[(size_t)(kk + 1) * H_ + n0 + t * 16 + row16];
                acc[t] = wmma_f32(a, bb, acc[t]);
            }
        }
        #pragma unroll
        for (int t = 0; t < 8; ++t) {
            int col = n0 + t * 16 + row16;
            #pragma unroll
            for (int i = 0; i < 8; ++i) {
                int r = i + hi * 8;
                size_t off = ((size_t)b * S_ + s0 + r) * H_ + col;
                out[off] = acc[t][i] + pself[r] * v_tok[off];
            }
        }
    }
}

extern "C" void kernel_launch(void* const* d_in, const int* in_sizes, int n_in,
                              void* d_out, int out_size, void* d_ws, size_t ws_size,
                              hipStream_t stream) {
    (void)in_sizes; (void)n_in; (void)out_size; (void)ws_size;
    const float* hidden = (const float*)d_in[0];
    const float* ext    = (const float*)d_in[1];
    const float* Wq     = (const float*)d_in[2];
    const float* bq     = (const float*)d_in[3];
    const float* Wk     = (const float*)d_in[4];
    const float* bk     = (const float*)d_in[5];
    const float* Wv     = (const float*)d_in[6];
    const float* bv     = (const float*)d_in[7];
    float* out = (float*)d_out;

    // Workspace (floats): q, k_tok, v_tok [B*S*H each]; k_ext, v_ext [B*E*H]
    const size_t nTok = (size_t)B_ * S_ * H_;
    const size_t nExt = (size_t)B_ * E_ * H_;
    float* q     = (float*)d_ws;
    float* k_tok = q + nTok;
    float* v_tok = k_tok + nTok;
    float* k_ext = v_tok + nTok;
    float* v_ext = k_ext + nExt;

    dim3 blk(256);
    dim3 gTok((B_ * S_) / 128, H_ / 64);   // (64, 16)
    dim3 gExt((B_ * E_) / 128, H_ / 64);   // (16, 16)

    proj_gemm_fused<3><<<gTok, blk, 0, stream>>>(hidden, Wq, Wk, Wv,
                                                 bq, bk, bv, q, k_tok, v_tok);
    proj_gemm_fused<2><<<gExt, blk, 0, stream>>>(ext, Wk, Wv, Wv,
                                                 bk, bv, bv, k_ext, v_ext, v_ext);

    attn_kernel<<<dim3(S_ / 16, B_), blk, 0, stream>>>(q, k_tok, v_tok,
                                                       k_ext, v_ext, out);
}